// SpconvModel_24369644438240
// MI455X (gfx1250) — compile-verified
//
#include <hip/hip_runtime.h>

typedef __attribute__((ext_vector_type(2))) float v2f;
typedef __attribute__((ext_vector_type(8))) float v8f;

#define BATCH 8
#define CIN   32
#define HH    512
#define WW    512
#define COUT  64
#define HW    (HH * WW)          // 262144 pixels per channel plane
#define WAVES_PER_BLOCK 8

// One wave handles one 16-pixel x 64-channel output tile.
// WMMA mapping: M = output channels (A = W^T tile), N = pixels (B = x tile), K = Cin.
// With N = pixels, B/C/D operands stripe pixels across lanes -> all global
// loads/stores are contiguous 64B runs per half-wave (perfectly coalesced in NCHW).
__global__ __launch_bounds__(256)
void spconv1x1_wmma_f32(const float* __restrict__ x,
                        const float* __restrict__ w,     // [CIN][COUT], co contiguous
                        const float* __restrict__ bias,  // [COUT]
                        float* __restrict__ out) {       // [B][COUT][H][W]
    const int lane = threadIdx.x & 31;
    const int wave = threadIdx.x >> 5;
    const int half = lane >> 4;     // 0: lanes 0-15, 1: lanes 16-31
    const int lm   = lane & 15;

    const int tile  = blockIdx.x * WAVES_PER_BLOCK + wave;
    const long g0   = (long)tile * 16;          // first pixel of tile (global)
    const int batch = (int)(g0 / HW);           // tiles never straddle a batch (HW % 16 == 0)
    const int pbase = (int)(g0 % HW);

    const float* xb = x   + (size_t)batch * CIN  * HW + pbase + lm;
    float*       ob = out + (size_t)batch * COUT * HW + pbase + lm;

    // ---- Load B operand: x tile, 16 pixels x 32 channels (16 VGPRs/lane) ----
    // B-matrix 4x16 f32 layout: VGPR v, lanes 0-15 -> (K=v, N=lane),
    // lanes 16-31 -> (K=v+2, N=lane-16). For k-step s, element v holds channel
    // ch = 4s + v + 2*half of pixel pbase+lm.
    v2f bt[8];
    int nz = 0;
    #pragma unroll
    for (int s = 0; s < 8; ++s) {
        #pragma unroll
        for (int v = 0; v < 2; ++v) {
            const int ch = 4 * s + v + 2 * half;
            const float val = xb[(size_t)ch * HW];
            bt[s][v] = val;
            nz |= (val != 0.0f) ? 1 : 0;
        }
    }
    // Lane l and lane l^16 together hold all 32 channels of pixel pbase+(l&15):
    // OR the two halves with ds_swizzle (group-of-32, xor_mask=0x10 -> 0x401F).
    nz |= __builtin_amdgcn_ds_swizzle(nz, 0x401F);
    const float maskf = nz ? 1.0f : 0.0f;

    // ---- 4 c_out tiles of 16 channels; K=32 in 8 steps of 4 ----
    #pragma unroll
    for (int t = 0; t < 4; ++t) {
        const int cobase = 16 * t;
        v8f acc = {};
        #pragma unroll
        for (int s = 0; s < 8; ++s) {
            // A-matrix 16x4 f32 layout: lane<16 -> (M=lane, K=4s+v),
            // lane>=16 -> (M=lane-16, K=4s+v+2). A[M][K] = W[K][cobase+M].
            v2f at;
            #pragma unroll
            for (int v = 0; v < 2; ++v) {
                const int k = 4 * s + v + 2 * half;
                at[v] = w[(size_t)k * COUT + cobase + lm];
            }
            // (neg_a, A, neg_b, B, c_mod, C, reuse_a, reuse_b)
            acc = __builtin_amdgcn_wmma_f32_16x16x4_f32(
                false, at, false, bt[s], (short)0, acc, false, false);
        }
        // ---- D layout: VGPR j, lanes 0-15 -> co=cobase+j, lanes 16-31 -> co=cobase+j+8;
        // pixel = pbase + lm. Stores are contiguous per half-wave. ----
        #pragma unroll
        for (int j = 0; j < 8; ++j) {
            const int co = cobase + j + 8 * half;
            const float y = (acc[j] + bias[co]) * maskf;
            ob[(size_t)co * HW] = y;
        }
    }
}

extern "C" void kernel_launch(void* const* d_in, const int* in_sizes, int n_in,
                              void* d_out, int out_size, void* d_ws, size_t ws_size,
                              hipStream_t stream) {
    const float* x  = (const float*)d_in[0];   // [8,32,512,512] f32
    const float* w  = (const float*)d_in[1];   // [32,64] f32
    const float* b  = (const float*)d_in[2];   // [64] f32
    float* out      = (float*)d_out;           // [8,64,512,512] f32

    const int totalTiles = (BATCH * HW) / 16;              // 131072
    const int blocks     = totalTiles / WAVES_PER_BLOCK;   // 16384
    spconv1x1_wmma_f32<<<blocks, 256, 0, stream>>>(x, w, b, out);
}